// RoiPoolingConv_90778428768349
// MI455X (gfx1250) — compile-verified
//
#include <hip/hip_runtime.h>

// ROI max-pool (pool_size=1): img (1,38,38,512) f32, rois (256,4) f32
// -> out (1,256,1,1,512) f32, flat [roi*512 + ch].
//
// CDNA5 design: one workgroup (8 waves) per ROI. Wave k owns channels
// [64k, 64k+64). Per ROI row, each wave issues ONE 2D Tensor-Data-Mover load
// (tile_dim0 = 64 channels, tile_dim1 = cols, dim0_stride = 512) that
// compacts its channel slice into a private LDS buffer. 3 buffers per wave,
// issue-2-ahead pipelining on TENSORcnt. No cross-wave barriers at all:
// every wave consumes exactly the data it DMA'd.

#define IMG_HW      38
#define IMG_C       512
#define CH_PER_WAVE 64          // 8 waves * 64 ch = 512
#define MAX_COLS    8           // w,h in [2,8) -> window spans <= 8 px
#define NBUF        3           // per-wave pipeline depth

typedef unsigned int u32x4 __attribute__((ext_vector_type(4)));
typedef unsigned int u32x8 __attribute__((ext_vector_type(8)));

// Low 32 bits of a flat LDS address == byte offset within the workgroup's
// LDS allocation.
__device__ __forceinline__ unsigned int lds_offset(const void* p) {
    return (unsigned int)(unsigned long long)(uintptr_t)p;
}

// Issue one 2D TDM load: tile_dim0 x tile_dim1 elements from gsrc
// (row stride = tensor_dim0_stride, both baked into g1) into LDS, compacted.
// 2D tensor -> 2-SGPR-group descriptor form (VADDR2/VADDR3 = NULL).
__device__ __forceinline__ void tdm_load_tile(const float* gsrc,
                                              unsigned int lds_addr,
                                              u32x8 g1) {
    unsigned long long ga = (unsigned long long)(uintptr_t)gsrc;
    u32x4 g0;
    g0[0] = 1u;                                    // count=1 (valid), user mode
    g0[1] = lds_addr;                              // lds_addr      [63:32]
    g0[2] = (unsigned int)ga;                      // global_addr   [95:64]
    g0[3] = (unsigned int)(ga >> 32) | (2u << 30); // addr hi + type=2 [127:126]
    asm volatile("tensor_load_to_lds %0, %1" :: "s"(g0), "s"(g1) : "memory");
}

// Waits as inline asm with "memory" clobber so the compiler cannot hoist the
// LDS consume loads above the tensor wait, nor sink prior ds_loads below the
// next DMA issue.
__device__ __forceinline__ void wait_tensorcnt2() { asm volatile("s_wait_tensorcnt 0x2" ::: "memory"); }
__device__ __forceinline__ void wait_tensorcnt1() { asm volatile("s_wait_tensorcnt 0x1" ::: "memory"); }
__device__ __forceinline__ void wait_tensorcnt0() { asm volatile("s_wait_tensorcnt 0x0" ::: "memory"); }
__device__ __forceinline__ void wait_dscnt0()     { asm volatile("s_wait_dscnt 0x0"     ::: "memory"); }

__global__ __launch_bounds__(256) void roi_maxpool_tdm_kernel(
    const float* __restrict__ img,   // (38,38,512)
    const float* __restrict__ rois,  // (nroi,4) = x,y,w,h
    float* __restrict__ out)         // (nroi,512)
{
    // 8 waves x 3 buffers x (8 cols * 64 ch) floats = 48 KB LDS.
    __shared__ float buf[8][NBUF][MAX_COLS * CH_PER_WAVE];

    const int roi  = blockIdx.x;
    const int t    = threadIdx.x;
    const int wave = t >> 5;     // 0..7  -> channel slice [64*wave, +64)
    const int lane = t & 31;     // owns channels 64*wave + 2*lane, +1

    // ROI params are block-uniform; compiler scalarizes (s_load / s_cvt).
    const float x = rois[roi * 4 + 0];
    const float y = rois[roi * 4 + 1];
    const float w = rois[roi * 4 + 2];
    const float h = rois[roi * 4 + 3];

    // Faithful to reference: trunc-to-int casts, x2 clamped by H, y2 by W
    // (reference quirk; harmless since H == W == 38).
    const int x1 = max(0, (int)x);
    const int y1 = max(0, (int)y);
    const int x2 = min(IMG_HW, (int)(x + w));
    const int y2 = min(IMG_HW, (int)(y + h));
    const int cols = x2 - x1;
    const int rows = y2 - y1;

    float a0 = -3.402823466e+38f;   // jnp.finfo(f32).min
    float a1 = -3.402823466e+38f;

    if (rows > 0 && cols > 0) {     // block-uniform branch
        // This wave's global base: (row y1, col x1, channel 64*wave).
        const float* base =
            img + ((size_t)y1 * IMG_HW + x1) * IMG_C + wave * CH_PER_WAVE;

        const unsigned int lb0 = lds_offset(&buf[wave][0][0]);
        const unsigned int lb1 = lds_offset(&buf[wave][1][0]);
        const unsigned int lb2 = lds_offset(&buf[wave][2][0]);

        // Loop-invariant descriptor group 1 (2D tile: 64 x cols).
        u32x8 g1;
        g1[0] = 2u << 16;                         // wg_mask=0, data_size=2 (4B)
        g1[1] = (unsigned)CH_PER_WAVE << 16;      // tensor_dim0 = 64   [79:48]
        g1[2] = (unsigned)IMG_HW << 16;           // tensor_dim1 = 38   [111:80]
        g1[3] = (unsigned)CH_PER_WAVE << 16;      // tile_dim0   = 64   [127:112]
        g1[4] = (unsigned)cols;                   // tile_dim1=cols, tile_dim2=0
        g1[5] = (unsigned)IMG_C;                  // tensor_dim0_stride = 512
        g1[6] = 0u;                               // stride0 hi / stride1 lo
        g1[7] = 0u;                               // stride1 hi (unused, 2D)

        // Prologue: up to 2 row-DMAs in flight per wave.
        tdm_load_tile(base, lb0, g1);
        if (rows > 1) tdm_load_tile(base + (IMG_HW * IMG_C), lb1, g1);

        for (int r = 0; r < rows; ++r) {
            const int b = r % NBUF;
            if (r + 2 < rows) {
                // Buffer (r+2)%3 was last read at iter r-1: drain our own
                // ds_loads before the TDM may overwrite it (TDM is unordered
                // vs this wave's LDS ops).
                wait_dscnt0();
                tdm_load_tile(base + (size_t)(r + 2) * (IMG_HW * IMG_C),
                              (r + 2) % NBUF == 0 ? lb0 :
                              (r + 2) % NBUF == 1 ? lb1 : lb2, g1);
                wait_tensorcnt2();   // 3 in flight -> oldest (row r) done
            } else if (r + 1 < rows) {
                wait_tensorcnt1();   // 2 in flight -> row r done
            } else {
                wait_tensorcnt0();   // last row
            }

            // Conflict-free LDS reads: lane l -> bytes (col*64 + 2l)*4,
            // 32 lanes x b64 span all 64 banks exactly once.
            const float2* rp = (const float2*)(&buf[wave][b][0]);
            for (int col = 0; col < cols; ++col) {
                float2 v = rp[col * (CH_PER_WAVE / 2) + lane];
                a0 = fmaxf(a0, v.x);
                a1 = fmaxf(a1, v.y);
            }
        }
    }

    // Channel pair (64*wave + 2*lane, +1).
    float2* op = (float2*)(out + (size_t)roi * IMG_C + wave * CH_PER_WAVE);
    op[lane] = make_float2(a0, a1);
}

extern "C" void kernel_launch(void* const* d_in, const int* in_sizes, int n_in,
                              void* d_out, int out_size, void* d_ws, size_t ws_size,
                              hipStream_t stream) {
    (void)n_in; (void)out_size; (void)d_ws; (void)ws_size;
    const float* img  = (const float*)d_in[0];   // (1,38,38,512) f32
    const float* rois = (const float*)d_in[1];   // (256,4) f32
    float* out = (float*)d_out;                  // (1,256,1,1,512) f32

    const int nroi = in_sizes[1] / 4;            // 256
    roi_maxpool_tdm_kernel<<<nroi, 256, 0, stream>>>(img, rois, out);
}